// SpatialTransformer_with_disp_12000138625393
// MI455X (gfx1250) — compile-verified
//
#include <hip/hip_runtime.h>
#include <stdint.h>

// ---- problem geometry (fixed by the reference) ----
#define Bn 2
#define Dn 160
#define Hn 192
#define Wn 224
#define SD (Hn * Wn)        // 43008  (z stride, elements)
#define SB (Dn * SD)        // 6881280 (batch stride, elements)
#define NVOX (Bn * SB)      // 13762560 voxels total
#define TPB 256
#define CHUNK 2048          // voxels per block: 13762560 / 2048 = 6720 exactly
#define VPT (CHUNK / TPB)   // 8 voxels per thread
#define FLOWF (CHUNK * 3)   // 6144 flow floats per block (24 KB LDS)

typedef __attribute__((ext_vector_type(4))) unsigned int u32x4;
typedef __attribute__((ext_vector_type(8))) int          i32x8;
typedef __attribute__((ext_vector_type(4))) int          i32x4;

#if __has_builtin(__builtin_amdgcn_tensor_load_to_lds)
#define HAVE_TDM 1
#else
#define HAVE_TDM 0
#endif

// Build a 1-row (degenerate 2D) Tensor DMA descriptor: n contiguous 4-byte
// elements at global address `ga` <-> LDS byte offset `lds_off`.
// Layout per CDNA5 ISA ch.7 §8 (D# groups 0 and 1).
__device__ __forceinline__ void tdm_desc_1d(u32x4& g0, i32x8& g1,
                                            unsigned lds_off,
                                            unsigned long long ga,
                                            unsigned n) {
    g0.x = 1u;                                    // count=1 (valid), user mode
    g0.y = lds_off;                               // lds_addr (bytes)
    g0.z = (unsigned)(ga & 0xffffffffull);        // global_addr[31:0]
    g0.w = (unsigned)((ga >> 32) & 0x1ffffffull)  // global_addr[56:32]
         | 0x80000000u;                           // type=2 ("image")
    g1[0] = (int)(2u << 16);                      // data_size=2 (4 bytes); mask=0
    g1[1] = (int)((n & 0xffffu) << 16);           // tensor_dim0[15:0]
    g1[2] = (int)((n >> 16) | (1u << 16));        // tensor_dim0[31:16] | tensor_dim1=1
    g1[3] = (int)((n & 0xffffu) << 16);           // tensor_dim1[31:16]=0 | tile_dim0
    g1[4] = 1;                                    // tile_dim1=1, tile_dim2=0
    g1[5] = (int)n;                               // tensor_dim0_stride[31:0]
    g1[6] = (int)((n & 0xffffu) << 16);           // stride0[47:32]=0 | stride1[15:0]
    g1[7] = (int)(n >> 16);                       // tensor_dim1_stride[47:16]
}

// Edge-clamped 1D linear weights, faithful to neurite interpn(fill_value=None)
__device__ __forceinline__ void axis_lin(float loc, int size,
                                         int& i0, int& i1, float& w0, float& w1) {
    float maxl = (float)(size - 1);
    float cl = fminf(fmaxf(loc, 0.0f), maxl);
    float l0 = fminf(fmaxf(floorf(loc), 0.0f), maxl);
    float l1 = fminf(l0 + 1.0f, maxl);
    i0 = (int)l0;
    i1 = (int)l1;
    w0 = l1 - cl;        // weight of low corner (neurite convention)
    w1 = 1.0f - w0;
}

__global__ void __launch_bounds__(TPB)
warp3d_tdm_kernel(const float* __restrict__ vol,
                  const float* __restrict__ flow,
                  float* __restrict__ out_warped,
                  float* __restrict__ out_flow) {
    __shared__ float lds_flow[FLOWF];

    const int base = blockIdx.x * CHUNK;            // first voxel of this block
    const unsigned lds_off = (unsigned)(uintptr_t)(void*)lds_flow; // LDS byte offset

#if HAVE_TDM
    // Wave 0: TDM DMA of this block's flow slice (6144 floats) into LDS.
    if (threadIdx.x < 32) {
        u32x4 g0; i32x8 g1;
        i32x4 z4 = {0, 0, 0, 0};
        tdm_desc_1d(g0, g1, lds_off,
                    (unsigned long long)(uintptr_t)(flow + (size_t)base * 3), FLOWF);
#if defined(__clang_major__) && (__clang_major__ >= 23)
        i32x8 z8 = {0, 0, 0, 0, 0, 0, 0, 0};
        __builtin_amdgcn_tensor_load_to_lds(g0, g1, z4, z4, z8, 0);
#else
        __builtin_amdgcn_tensor_load_to_lds(g0, g1, z4, z4, 0);
#endif
        __builtin_amdgcn_s_wait_tensorcnt(0);       // data landed in LDS
    }
    __syncthreads();
    // Wave 0: immediately stream the same LDS bytes out as the flow passthrough
    // output (reference returns flow unchanged) while all waves compute.
    if (threadIdx.x < 32) {
        u32x4 g0; i32x8 g1;
        i32x4 z4 = {0, 0, 0, 0};
        tdm_desc_1d(g0, g1, lds_off,
                    (unsigned long long)(uintptr_t)(out_flow + (size_t)base * 3), FLOWF);
#if defined(__clang_major__) && (__clang_major__ >= 23)
        i32x8 z8 = {0, 0, 0, 0, 0, 0, 0, 0};
        __builtin_amdgcn_tensor_store_from_lds(g0, g1, z4, z4, z8, 0);
#else
        __builtin_amdgcn_tensor_store_from_lds(g0, g1, z4, z4, 0);
#endif
    }
#else
    // Fallback (toolchain without TDM builtins): cooperative staged copy.
    for (int i = threadIdx.x; i < FLOWF; i += TPB) {
        float v = flow[(size_t)base * 3 + i];
        lds_flow[i] = v;
        out_flow[(size_t)base * 3 + i] = v;
    }
    __syncthreads();
#endif

    // ---- trilinear warp: 8 voxels per thread, flow read from LDS ----
    for (int k = 0; k < VPT; ++k) {
        const int l = threadIdx.x + k * TPB;        // local voxel
        const int g = base + l;                      // global voxel

        const float fz = lds_flow[3 * l + 0];        // flow[...,0] -> dim D
        const float fy = lds_flow[3 * l + 1];        // flow[...,1] -> dim H
        const float fx = lds_flow[3 * l + 2];        // flow[...,2] -> dim W

        int x = g % Wn;
        int t = g / Wn;
        int y = t % Hn;  t /= Hn;
        int z = t % Dn;
        int b = t / Dn;

        int z0, z1, y0, y1, x0, x1;
        float wz0, wz1, wy0, wy1, wx0, wx1;
        axis_lin((float)z + fz, Dn, z0, z1, wz0, wz1);
        axis_lin((float)y + fy, Hn, y0, y1, wy0, wy1);
        axis_lin((float)x + fx, Wn, x0, x1, wx0, wx1);

        const float* vb = vol + (size_t)b * SB;
        const int oz0 = z0 * SD, oz1 = z1 * SD;
        const int oy0 = y0 * Wn, oy1 = y1 * Wn;

        const float v000 = vb[oz0 + oy0 + x0];
        const float v001 = vb[oz0 + oy0 + x1];
        const float v010 = vb[oz0 + oy1 + x0];
        const float v011 = vb[oz0 + oy1 + x1];
        const float v100 = vb[oz1 + oy0 + x0];
        const float v101 = vb[oz1 + oy0 + x1];
        const float v110 = vb[oz1 + oy1 + x0];
        const float v111 = vb[oz1 + oy1 + x1];

        const float r =
            wz0 * (wy0 * (wx0 * v000 + wx1 * v001) +
                   wy1 * (wx0 * v010 + wx1 * v011)) +
            wz1 * (wy0 * (wx0 * v100 + wx1 * v101) +
                   wy1 * (wx0 * v110 + wx1 * v111));

        out_warped[g] = r;
    }
}

extern "C" void kernel_launch(void* const* d_in, const int* in_sizes, int n_in,
                              void* d_out, int out_size, void* d_ws, size_t ws_size,
                              hipStream_t stream) {
    (void)in_sizes; (void)n_in; (void)out_size; (void)d_ws; (void)ws_size;
    const float* vol  = (const float*)d_in[0];   // [2,160,192,224,1] f32
    const float* flow = (const float*)d_in[1];   // [2,160,192,224,3] f32
    float* out_warped = (float*)d_out;           // first NVOX floats
    float* out_flow   = out_warped + (size_t)NVOX; // then 3*NVOX floats (flow copy)

    dim3 grid(NVOX / CHUNK);                     // 6720 blocks, exact
    warp3d_tdm_kernel<<<grid, TPB, 0, stream>>>(vol, flow, out_warped, out_flow);
}